// TreeEmbeddingModel_35167192220011
// MI455X (gfx1250) — compile-verified
//
#include <hip/hip_runtime.h>

// Per-batch row gather: out[b, :] = emb[b, idx[b], :]
//   emb: [B, C, H] fp32, idx: [B] int, out: [B, H] fp32
//
// Memory-bound (0 FLOPs, ~8 MB useful traffic -> ~0.35 us at 23.3 TB/s).
// Strategy:
//   - one block per batch row, H/4 lanes, one 128-bit (float4) op per lane
//   - non-temporal load AND store (streaming, zero reuse; 512 MB input
//     working set >> 192 MB L2 -> keep it out of the cache hierarchy)
//   - wave-uniform index load -> scalar s_load (KMcnt path)
//
// Native clang vector type so __builtin_nontemporal_* accepts it and lowers
// to global_load_b128 / global_store_b128 with th:NT.
typedef float v4f __attribute__((ext_vector_type(4)));

__global__ __launch_bounds__(256) void tree_embed_gather_kernel(
    const float* __restrict__ emb,   // [B, C, H]
    const int*   __restrict__ idx,   // [B]
    float*       __restrict__ out,   // [B, H]
    int C, int H4)                   // H4 = H/4 (float4 units per row)
{
    const int b = blockIdx.x;
    // Uniform across the block -> scalar load.
    const int c = idx[b];

    const v4f* __restrict__ src =
        reinterpret_cast<const v4f*>(emb) + ((size_t)b * C + (size_t)c) * H4;
    v4f* __restrict__ dst =
        reinterpret_cast<v4f*>(out) + (size_t)b * H4;

    // H=256 -> H4=64 -> exactly one iteration per lane with blockDim.x=64.
    for (int col = threadIdx.x; col < H4; col += blockDim.x) {
        v4f v = __builtin_nontemporal_load(src + col);   // global_load_b128  th:NT
        __builtin_nontemporal_store(v, dst + col);       // global_store_b128 th:NT
    }
}

extern "C" void kernel_launch(void* const* d_in, const int* in_sizes, int n_in,
                              void* d_out, int out_size, void* d_ws, size_t ws_size,
                              hipStream_t stream) {
    (void)n_in; (void)d_ws; (void)ws_size;

    const float* emb = (const float*)d_in[0];   // [B, C, H] fp32
    const int*   idx = (const int*)d_in[1];     // [B] (integer -> int32 per harness)
    float*       out = (float*)d_out;           // [B, H] fp32

    const int B = in_sizes[1];                  // 4096
    const int H = out_size / B;                 // 256
    const long long total = (long long)in_sizes[0];
    const int C = (int)(total / ((long long)B * (long long)H));  // 128

    const int H4 = H / 4;                       // 64 float4 per row
    int tpb = H4;                                // 64 threads = 2 wave32
    if (tpb > 256) tpb = 256;
    if (tpb < 32)  tpb = 32;

    tree_embed_gather_kernel<<<B, tpb, 0, stream>>>(emb, idx, out, C, H4);
}